// SLAHead_8366596292987
// MI455X (gfx1250) — compile-verified
//
#include <hip/hip_runtime.h>
#include <hip/hip_bf16.h>

// ---------------------------------------------------------------------------
// SLAHead for MI455X (gfx1250, wave32).
//   - H_proj once on V_WMMA_F32_16X16X4_F32, stored transposed [b,h,l].
//   - Per step (T=501): 3 kernels
//       k_attn     : hproj + tanh-dot + softmax + context.  Uses hardware
//                    V_TANH_F32 (dominant cost: 8.4M tanh/step), 512-thread
//                    blocks with split-h reduction for 2x parallelism,
//                    b128-vectorized context dot.
//       k_gru_gemm : gi/gh GEMMs on fp32 WMMA (8 waves/block).
//       k_fused    : GRU gates + heads + argmax + direct softmax/sigmoid
//                    writes into d_out.
//   Working set (x 67MB + Hp 33.5MB + weights) < 192MB L2 -> L2-resident loop.
// ---------------------------------------------------------------------------

#define BSZ  128
#define CIN  512
#define LSP  256   // FH*FW
#define HIDN 256
#define OUTC 30
#define LOCN 4
#define TSTP 501
#define G3   768   // 3*HIDN
#define XIN  542   // CIN + OUTC

typedef __attribute__((ext_vector_type(2))) float v2f;
typedef __attribute__((ext_vector_type(8))) float v8f;

__device__ __forceinline__ v8f wmma_f32_16x16x4(v2f a, v2f b, v8f c) {
  // 8 args: (neg_a, A, neg_b, B, c_mod, C, reuse_a, reuse_b)
  return __builtin_amdgcn_wmma_f32_16x16x4_f32(false, a, false, b, (short)0, c,
                                               false, false);
}

// gfx1250 hardware transcendental tanh (V_TANH_F32): single trans-unit op
// instead of the multi-instruction OCML polynomial.
__device__ __forceinline__ float fast_tanh(float x) {
#if __has_builtin(__builtin_amdgcn_tanhf)
  return __builtin_amdgcn_tanhf(x);
#else
  float r;
  asm("v_tanh_f32 %0, %1" : "=v"(r) : "v"(x));
  return r;
#endif
}

__device__ __forceinline__ float fast_sigmoid(float x) {
  return __builtin_amdgcn_rcpf(1.f + __expf(-x));  // v_exp + v_rcp
}

// ---------------------------------------------------------------------------
// One-time: transpose small square weights (for coalesced per-step VALU dots),
// zero hidden state and pre_chars.
// ---------------------------------------------------------------------------
__global__ void k_setup(const float* __restrict__ w_h2h,
                        const float* __restrict__ w_sg1,
                        const float* __restrict__ w_lg1,
                        float* __restrict__ w_h2h_t,
                        float* __restrict__ w_sg1_t,
                        float* __restrict__ w_lg1_t,
                        float* __restrict__ hidden,
                        int* __restrict__ pre_chars) {
  int idx = blockIdx.x * blockDim.x + threadIdx.x;
  if (idx < HIDN * HIDN) {
    int r = idx / HIDN, c = idx % HIDN;
    w_h2h_t[c * HIDN + r] = w_h2h[idx];
    w_sg1_t[c * HIDN + r] = w_sg1[idx];
    w_lg1_t[c * HIDN + r] = w_lg1[idx];
  }
  if (idx < BSZ * HIDN) hidden[idx] = 0.f;
  if (idx < BSZ) pre_chars[idx] = 0;
}

// ---------------------------------------------------------------------------
// One-time: Hp[b][h][l] = sum_c x[b][c][l] * w_i2h[h][c]   (fp32 WMMA)
// M<->h, N<->l, K<->c  => B-matrix loads and C-tile stores are coalesced.
// Grid: (L/16, HID/16/8, B), 256 threads (8 waves, one 16x16 tile per wave).
// ---------------------------------------------------------------------------
__global__ void __launch_bounds__(256)
k_hproj(const float* __restrict__ x, const float* __restrict__ w_i2h,
        float* __restrict__ Hp) {
  const int lane = threadIdx.x & 31;
  const int wave = threadIdx.x >> 5;
  const int kh   = lane >> 4;    // half-wave id (K sub-pair select)
  const int id16 = lane & 15;
  const int lt = blockIdx.x, b = blockIdx.z;
  const int ht = blockIdx.y * 8 + wave;
  const int h0 = ht * 16, l0 = lt * 16;

  const float* arow = w_i2h + (size_t)(h0 + id16) * CIN;      // A[m=id16][k]
  const float* bcol = x + (size_t)b * CIN * LSP + l0 + id16;  // B[k][n=id16]

  v8f acc = {};
  for (int k = 0; k < CIN; k += 4) {
    const int ka = k + 2 * kh;
    v2f a  = { arow[ka], arow[ka + 1] };
    v2f bb = { bcol[(size_t)ka * LSP], bcol[(size_t)(ka + 1) * LSP] };
    acc = wmma_f32_16x16x4(a, bb, acc);
  }
  float* crow = Hp + ((size_t)b * HIDN + h0) * LSP + l0;
#pragma unroll
  for (int r = 0; r < 8; ++r) {
    const int m = r + 8 * kh;                  // C: lanes0-15 M=r, lanes16-31 M=8+r
    crow[(size_t)m * LSP + id16] = acc[r];     // coalesced over id16 (=l)
  }
}

// ---------------------------------------------------------------------------
// Per step: fused hidden-proj + attention + softmax + context.
// One block of 512 threads per batch element; thread (hh,l) splits the
// h-reduction in half to shorten the serial critical path.
// ---------------------------------------------------------------------------
__global__ void __launch_bounds__(512)
k_attn(const float* __restrict__ Hp, const float* __restrict__ x,
       const float* __restrict__ hidden, const float* __restrict__ w_h2h_t,
       const float* __restrict__ w_score, float* __restrict__ ctx) {
  __shared__ float h_s[HIDN];
  __shared__ float ws_s[HIDN];
  __shared__ float hp_part[2][HIDN];
  __shared__ float hp_s[HIDN];
  __shared__ float e_part[2][LSP];
  __shared__ __align__(16) float a_s[LSP];
  __shared__ float red_m[8];
  __shared__ float red_sum[8];

  const int b = blockIdx.x;
  const int t = threadIdx.x;        // 0..511
  const int l = t & (LSP - 1);      // 0..255
  const int hh = t >> 8;            // which half of the h range
  const int wave = t >> 5, laneid = t & 31;

  if (t < HIDN) {
    h_s[t]  = hidden[b * HIDN + t];
    ws_s[t] = w_score[t];
  }
  __syncthreads();

  // hp[h] = sum_k h[k] * w_h2h[h][k]   (split over k halves, coalesced reads)
  {
    float acc = 0.f;
    const int k0 = hh * (HIDN / 2);
    for (int k = k0; k < k0 + HIDN / 2; ++k)
      acc = fmaf(h_s[k], w_h2h_t[k * HIDN + l], acc);
    hp_part[hh][l] = acc;
  }
  __syncthreads();
  if (t < HIDN) hp_s[t] = hp_part[0][t] + hp_part[1][t];
  __syncthreads();

  // e[l] = sum_h tanh(Hp[b][h][l] + hp[h]) * ws[h]   (split over h halves)
  {
    const float* hprow =
        Hp + (size_t)b * HIDN * LSP + (size_t)hh * (HIDN / 2) * LSP + l;
    const float* hps = hp_s + hh * (HIDN / 2);
    const float* wss = ws_s + hh * (HIDN / 2);
    float e = 0.f;
    for (int h = 0; h < HIDN / 2; ++h)
      e = fmaf(fast_tanh(hprow[(size_t)h * LSP] + hps[h]), wss[h], e);
    e_part[hh][l] = e;
  }
  __syncthreads();

  // block softmax over the 256 combined scores (first 8 waves only)
  float e = 0.f, ex = 0.f;
  if (t < LSP) {
    e = e_part[0][t] + e_part[1][t];
    float m = e;
    for (int off = 16; off > 0; off >>= 1) m = fmaxf(m, __shfl_xor(m, off, 32));
    if (laneid == 0) red_m[wave] = m;
  }
  __syncthreads();
  if (t < LSP) {
    float mall = red_m[0];
#pragma unroll
    for (int i = 1; i < 8; ++i) mall = fmaxf(mall, red_m[i]);
    ex = __expf(e - mall);
    float s = ex;
    for (int off = 16; off > 0; off >>= 1) s += __shfl_xor(s, off, 32);
    if (laneid == 0) red_sum[wave] = s;
  }
  __syncthreads();
  if (t < LSP) {
    float sall = red_sum[0];
#pragma unroll
    for (int i = 1; i < 8; ++i) sall += red_sum[i];
    a_s[t] = ex * __builtin_amdgcn_rcpf(sall);
  }
  __syncthreads();

  // context[b][c] = sum_l alpha[l] * x[b][c][l]
  // wave-per-row, b128 vectorized, 16 waves
  const float* xb = x + (size_t)b * CIN * LSP;
  const float4* a4 = (const float4*)a_s;
  for (int c = wave; c < CIN; c += 16) {
    const float4* xr = (const float4*)(xb + (size_t)c * LSP);
    float acc = 0.f;
    for (int j = laneid; j < LSP / 4; j += 32) {
      const float4 xv = xr[j];
      const float4 av = a4[j];
      acc = fmaf(xv.x, av.x, acc);
      acc = fmaf(xv.y, av.y, acc);
      acc = fmaf(xv.z, av.z, acc);
      acc = fmaf(xv.w, av.w, acc);
    }
    for (int off = 16; off > 0; off >>= 1) acc += __shfl_xor(acc, off, 32);
    if (laneid == 0) ctx[b * CIN + c] = acc;
  }
}

// ---------------------------------------------------------------------------
// Per step: gi = ctx @ w_ih[:, :512].T  and  gh = hidden @ w_hh.T  (fp32 WMMA)
// Grid (8 M-tiles, 6 N-groups), 8 waves per block, one 16x16 tile per wave.
// ---------------------------------------------------------------------------
__global__ void __launch_bounds__(256)
k_gru_gemm(const float* __restrict__ ctx, const float* __restrict__ hidden,
           const float* __restrict__ w_ih, const float* __restrict__ w_hh,
           float* __restrict__ gi, float* __restrict__ gh) {
  const int lane = threadIdx.x & 31;
  const int wave = threadIdx.x >> 5;
  const int kh = lane >> 4, id16 = lane & 15;
  const int b0 = blockIdx.x * 16;
  const int j0 = (blockIdx.y * 8 + wave) * 16;

  {  // gi tile: K = CIN (one-hot column folded in later)
    const float* arow = ctx + (size_t)(b0 + id16) * CIN;
    const float* brow = w_ih + (size_t)(j0 + id16) * XIN;
    v8f acc = {};
    for (int k = 0; k < CIN; k += 4) {
      const int ka = k + 2 * kh;
      v2f a  = { arow[ka], arow[ka + 1] };
      v2f bb = { brow[ka], brow[ka + 1] };
      acc = wmma_f32_16x16x4(a, bb, acc);
    }
    float* out = gi + (size_t)b0 * G3 + j0;
#pragma unroll
    for (int r = 0; r < 8; ++r) out[(size_t)(r + 8 * kh) * G3 + id16] = acc[r];
  }
  {  // gh tile: K = HIDN
    const float* arow = hidden + (size_t)(b0 + id16) * HIDN;
    const float* brow = w_hh + (size_t)(j0 + id16) * HIDN;
    v8f acc = {};
    for (int k = 0; k < HIDN; k += 4) {
      const int ka = k + 2 * kh;
      v2f a  = { arow[ka], arow[ka + 1] };
      v2f bb = { brow[ka], brow[ka + 1] };
      acc = wmma_f32_16x16x4(a, bb, acc);
    }
    float* out = gh + (size_t)b0 * G3 + j0;
#pragma unroll
    for (int r = 0; r < 8; ++r) out[(size_t)(r + 8 * kh) * G3 + id16] = acc[r];
  }
}

// ---------------------------------------------------------------------------
// Per step: GRU gates -> h_new; heads MLPs; per-row softmax/argmax/sigmoid
// written straight into d_out.  One block per batch element, thread j = unit.
// ---------------------------------------------------------------------------
__global__ void __launch_bounds__(256)
k_fused(const float* __restrict__ gi, const float* __restrict__ gh,
        const float* __restrict__ w_ih, const float* __restrict__ b_ih,
        const float* __restrict__ b_hh, float* __restrict__ hidden,
        int* __restrict__ pre_chars,
        const float* __restrict__ w_sg1_t, const float* __restrict__ b_sg1,
        const float* __restrict__ w_sg2,  const float* __restrict__ b_sg2,
        const float* __restrict__ w_lg1_t, const float* __restrict__ b_lg1,
        const float* __restrict__ w_lg2,  const float* __restrict__ b_lg2,
        float* __restrict__ out_struct, float* __restrict__ out_loc,
        int t_step) {
  __shared__ float h_s[HIDN];
  __shared__ float sg_s[HIDN];
  __shared__ float lg_s[HIDN];
  __shared__ float logit_s[32];

  const int b = blockIdx.x;
  const int j = threadIdx.x;
  const int pc = pre_chars[b];  // read old value before any write

  // GRU gates (one-hot contribution == column CIN+pc of w_ih)
  const float ir  = gi[b * G3 + j]            + w_ih[(size_t)j * XIN + CIN + pc]              + b_ih[j];
  const float iz  = gi[b * G3 + HIDN + j]     + w_ih[(size_t)(HIDN + j) * XIN + CIN + pc]     + b_ih[HIDN + j];
  const float inn = gi[b * G3 + 2 * HIDN + j] + w_ih[(size_t)(2 * HIDN + j) * XIN + CIN + pc] + b_ih[2 * HIDN + j];
  const float hr = gh[b * G3 + j]            + b_hh[j];
  const float hz = gh[b * G3 + HIDN + j]     + b_hh[HIDN + j];
  const float hn = gh[b * G3 + 2 * HIDN + j] + b_hh[2 * HIDN + j];

  const float r = fast_sigmoid(ir + hr);
  const float z = fast_sigmoid(iz + hz);
  const float n = fast_tanh(inn + r * hn);
  const float h_new = (1.f - z) * n + z * hidden[b * HIDN + j];
  hidden[b * HIDN + j] = h_new;
  h_s[j] = h_new;
  __syncthreads();

  // head hidden layers (coalesced via transposed weights)
  float s1 = 0.f, l1 = 0.f;
  for (int k = 0; k < HIDN; ++k) {
    const float hv = h_s[k];
    s1 = fmaf(hv, w_sg1_t[k * HIDN + j], s1);
    l1 = fmaf(hv, w_lg1_t[k * HIDN + j], l1);
  }
  sg_s[j] = s1 + b_sg1[j];
  lg_s[j] = l1 + b_lg1[j];
  __syncthreads();

  if (j < OUTC) {  // structure logits
    float acc = b_sg2[j];
    for (int k = 0; k < HIDN; ++k) acc = fmaf(sg_s[k], w_sg2[j * HIDN + k], acc);
    logit_s[j] = acc;
  }
  if (j >= 32 && j < 32 + LOCN) {  // loc head, sigmoid, straight to d_out
    const int o = j - 32;
    float acc = b_lg2[o];
    for (int k = 0; k < HIDN; ++k) acc = fmaf(lg_s[k], w_lg2[o * HIDN + k], acc);
    out_loc[((size_t)b * TSTP + t_step) * LOCN + o] = fast_sigmoid(acc);
  }
  __syncthreads();

  if (j == 0) {  // softmax + argmax over 30 logits, straight to d_out
    float mx = logit_s[0];
    int am = 0;
    for (int o = 1; o < OUTC; ++o)
      if (logit_s[o] > mx) { mx = logit_s[o]; am = o; }
    float exs[OUTC], s = 0.f;
    for (int o = 0; o < OUTC; ++o) { exs[o] = __expf(logit_s[o] - mx); s += exs[o]; }
    const float inv = __builtin_amdgcn_rcpf(s);
    float* op = out_struct + ((size_t)b * TSTP + t_step) * OUTC;
    for (int o = 0; o < OUTC; ++o) op[o] = exs[o] * inv;
    pre_chars[b] = am;
  }
}

// ---------------------------------------------------------------------------
extern "C" void kernel_launch(void* const* d_in, const int* in_sizes, int n_in,
                              void* d_out, int out_size, void* d_ws,
                              size_t ws_size, hipStream_t stream) {
  (void)in_sizes; (void)n_in; (void)out_size; (void)ws_size;
  const float* x       = (const float*)d_in[0];
  const float* w_i2h   = (const float*)d_in[1];
  const float* w_h2h   = (const float*)d_in[2];
  const float* w_score = (const float*)d_in[3];
  const float* w_ih    = (const float*)d_in[4];
  const float* w_hh    = (const float*)d_in[5];
  const float* b_ih    = (const float*)d_in[6];
  const float* b_hh    = (const float*)d_in[7];
  const float* w_sg1   = (const float*)d_in[8];
  const float* b_sg1   = (const float*)d_in[9];
  const float* w_sg2   = (const float*)d_in[10];
  const float* b_sg2   = (const float*)d_in[11];
  const float* w_lg1   = (const float*)d_in[12];
  const float* b_lg1   = (const float*)d_in[13];
  const float* w_lg2   = (const float*)d_in[14];
  const float* b_lg2   = (const float*)d_in[15];

  // workspace carve-up (~35.5 MB of f32)
  float* ws = (float*)d_ws;
  size_t o = 0;
  float* Hp       = ws + o; o += (size_t)BSZ * HIDN * LSP;
  float* w_h2h_t  = ws + o; o += (size_t)HIDN * HIDN;
  float* w_sg1_t  = ws + o; o += (size_t)HIDN * HIDN;
  float* w_lg1_t  = ws + o; o += (size_t)HIDN * HIDN;
  float* hidden   = ws + o; o += (size_t)BSZ * HIDN;
  float* ctx      = ws + o; o += (size_t)BSZ * CIN;
  float* gi       = ws + o; o += (size_t)BSZ * G3;
  float* gh       = ws + o; o += (size_t)BSZ * G3;
  int*   pre      = (int*)(ws + o);

  float* out_struct = (float*)d_out;
  float* out_loc    = out_struct + (size_t)BSZ * TSTP * OUTC;

  k_setup<<<(HIDN * HIDN + 255) / 256, 256, 0, stream>>>(
      w_h2h, w_sg1, w_lg1, w_h2h_t, w_sg1_t, w_lg1_t, hidden, pre);
  k_hproj<<<dim3(LSP / 16, HIDN / 16 / 8, BSZ), 256, 0, stream>>>(x, w_i2h, Hp);

  for (int t = 0; t < TSTP; ++t) {
    k_attn<<<BSZ, 512, 0, stream>>>(Hp, x, hidden, w_h2h_t, w_score, ctx);
    k_gru_gemm<<<dim3(BSZ / 16, G3 / 16 / 8), 256, 0, stream>>>(
        ctx, hidden, w_ih, w_hh, gi, gh);
    k_fused<<<BSZ, 256, 0, stream>>>(gi, gh, w_ih, b_ih, b_hh, hidden, pre,
                                     w_sg1_t, b_sg1, w_sg2, b_sg2, w_lg1_t,
                                     b_lg1, w_lg2, b_lg2, out_struct, out_loc,
                                     t);
  }
}